// SimpleTreeLayer_44444321579224
// MI455X (gfx1250) — compile-verified
//
#include <hip/hip_runtime.h>
#include <math.h>

// ---------------- problem constants (from reference) ----------------
constexpr int DIMN   = 256;   // feature dim (also GEMM N)
constexpr int NNODES = 2000;
constexpr int BATCH  = 64;
constexpr int MAXL   = 40;    // max tree levels supported (tree height ~12-18 expected)

typedef __attribute__((ext_vector_type(2))) float v2f;
typedef __attribute__((ext_vector_type(8))) float v8f;

// ---------------------------------------------------------------
// Kernel 1: build level schedule on device.
// depth[i] = 0 for leaves, 1+max(child depths) otherwise. Children idx > parent
// idx, so parallel relaxation converges monotonically in <= height passes.
// Buckets node ids by depth into level_nodes[], with counts/offsets.
// ---------------------------------------------------------------
__global__ __launch_bounds__(256) void plan_kernel(const int* __restrict__ structure,
                                                   int* __restrict__ level_count,
                                                   int* __restrict__ level_offset,
                                                   int* __restrict__ level_nodes) {
  __shared__ int ch_s[NNODES];
  __shared__ int ch_e[NNODES];
  __shared__ int depth_s[NNODES];
  __shared__ int cnt_s[MAXL];
  __shared__ int cur_s[MAXL];
  __shared__ int off_s[MAXL + 1];
  const int tid = threadIdx.x;

  for (int i = tid; i < NNODES; i += 256) {
    ch_s[i] = structure[2 * i];      // batch 0 of [B,N,2] (shared across batch)
    ch_e[i] = structure[2 * i + 1];
    depth_s[i] = 0;
  }
  if (tid < MAXL) { cnt_s[tid] = 0; cur_s[tid] = 0; }
  __syncthreads();

  for (int pass = 0; pass < MAXL; ++pass) {
    for (int i = tid; i < NNODES; i += 256) {
      int s = ch_s[i];
      if (s >= 0) {
        int e = ch_e[i];
        int mx = 0;
        for (int c = s; c < e; ++c) mx = (depth_s[c] > mx) ? depth_s[c] : mx;
        depth_s[i] = mx + 1;   // monotone non-decreasing across passes
      }
    }
    __syncthreads();
  }

  for (int i = tid; i < NNODES; i += 256) {
    int d = depth_s[i]; if (d > MAXL - 1) d = MAXL - 1;
    atomicAdd(&cnt_s[d], 1);
  }
  __syncthreads();
  if (tid == 0) {
    int acc = 0;
    for (int d = 0; d < MAXL; ++d) { off_s[d] = acc; acc += cnt_s[d]; }
    off_s[MAXL] = acc;
  }
  __syncthreads();
  for (int i = tid; i < NNODES; i += 256) {
    int d = depth_s[i]; if (d > MAXL - 1) d = MAXL - 1;
    int pos = off_s[d] + atomicAdd(&cur_s[d], 1);
    level_nodes[pos] = i;
  }
  __syncthreads();
  for (int d = tid; d < MAXL; d += 256) {
    level_count[d]  = cnt_s[d];
    level_offset[d] = off_s[d];
  }
}

// ---------------------------------------------------------------
// Kernel 2: build WMMA-fragment-layout weights.
// Fragment layout: element (k, n) at Wt[((k>>1)*256 + n)*2 + (k&1)], so the
// K-pair a lane needs (W[kk][n], W[kk+1][n]) is a single aligned b64, and
// lanes 0..15 (consecutive n) are fully coalesced.
//   WtInner (512x256): Wc[k] = W_inner[k] + W_inner[k+256] (k<256)
//                      Wc[k] = W_inner[k+256]              (k>=256)
//     (folds h = [c_mean, c_mean, c_min] -> [c_mean, c_min], K 768->512)
//   WtLeaf  (256x256): W_in
// ---------------------------------------------------------------
__global__ void prep_w_kernel(const float* __restrict__ Wi,
                              const float* __restrict__ Win,
                              float* __restrict__ WtInner,
                              float* __restrict__ WtLeaf) {
  int idx = blockIdx.x * blockDim.x + threadIdx.x;
  if (idx < 512 * 256) {
    int k = idx >> 8, n = idx & 255;
    float v = (k < 256) ? Wi[k * 256 + n] + Wi[(k + 256) * 256 + n]
                        : Wi[(k + 256) * 256 + n];
    WtInner[(((k >> 1) * 256) + n) * 2 + (k & 1)] = v;
  } else if (idx < 768 * 256) {
    int j = idx - 512 * 256;
    int k = j >> 8, n = j & 255;
    WtLeaf[(((k >> 1) * 256) + n) * 2 + (k & 1)] = Win[k * 256 + n];
  }
}

// ---------------------------------------------------------------
// Kernel 3: per-level fused pool + GEMM + sigmoid, fp32 WMMA.
// Block = (node within level) x (half of batch: 32 rows).
// A-tile 32x512 (inner) / 32x256 (leaf) staged in 64 KB LDS.
// 4 waves: wave w owns ALL 32 rows x cols [w*64, w*64+64): 2 row-tiles x
// 4 col-tiles = 8 v8f accumulators; each B fragment (global b64 from the
// fragment-layout W) is reused across both row-tiles.
// K stepped by 4 via v_wmma_f32_16x16x4_f32.
// state layout: [node][batch][256].
// ---------------------------------------------------------------
__global__ __launch_bounds__(128) void tree_gemm_kernel(
    int lvl,
    const float* __restrict__ features,   // [B, N, 256]
    const float* __restrict__ W,          // fragment layout; lvl==0: WtLeaf else WtInner
    const float* __restrict__ bias,       // [256]
    const int*   __restrict__ structure,  // [N,2] (batch 0)
    const int*   __restrict__ level_count,
    const int*   __restrict__ level_offset,
    const int*   __restrict__ level_nodes,
    float*       __restrict__ state)      // [N, B, 256], read (children) + write (node)
{
  __shared__ float As[32 * 512];          // 65536 bytes
  const int cnt = level_count[lvl];
  const int node_i = blockIdx.x >> 1;
  if (node_i >= cnt) return;              // block-uniform exit (EXEC stays all-ones after)
  const int b_off = (blockIdx.x & 1) * 32;
  const int node = level_nodes[level_offset[lvl] + node_i];
  const int tid = threadIdx.x;

  int Kdim, lda;
  if (lvl == 0) {
    Kdim = 256; lda = 256;
    for (int idx = tid; idx < 32 * 256; idx += 128) {
      int bl = idx >> 8, k = idx & 255;
      int b = b_off + bl;
      As[bl * 256 + k] = features[((size_t)b * NNODES + node) * DIMN + k];
    }
  } else {
    Kdim = 512; lda = 512;
    const int s = structure[2 * node];
    const int e = structure[2 * node + 1];
    const float inv = 1.0f / (float)(e - s + 1);
    for (int idx = tid; idx < 32 * 256; idx += 128) {
      int bl = idx >> 8, k = idx & 255;
      int b = b_off + bl;
      float f = features[((size_t)b * NNODES + node) * DIMN + k];
      float acc = f, mn = f;
      for (int c = s; c < e; ++c) {
        float v = state[((size_t)c * BATCH + b) * DIMN + k];
        acc += v;
        mn = fminf(mn, v);
      }
      As[bl * 512 + k]       = acc * inv;  // c_mean
      As[bl * 512 + 256 + k] = mn;         // c_min
    }
  }
  __syncthreads();

  const int lane = tid & 31;
  const int wave = tid >> 5;
  const int n0 = wave * 64;               // this wave's 64-column strip
  const int lm = lane & 15;
  const int lk = (lane >> 4) << 1;        // lanes 16-31 carry K+2,K+3 (ISA A/B layout)

  v8f acc[2][4];
  {
    v8f z = {};
    #pragma unroll
    for (int rt = 0; rt < 2; ++rt)
      #pragma unroll
      for (int t = 0; t < 4; ++t) acc[rt][t] = z;
  }

  for (int k0 = 0; k0 < Kdim; k0 += 4) {
    const int kk = k0 + lk;
    const float* ap = &As[lm * lda + kk];
    v2f a0, a1;
    a0.x = ap[0];
    a0.y = ap[1];
    a1.x = ap[16 * lda];
    a1.y = ap[16 * lda + 1];
    // fragment-layout W: float offset of (kk, n) pair = kk*256 + 2*n
    const float* wp = W + kk * DIMN + 2 * (n0 + lm);
    #pragma unroll
    for (int t = 0; t < 4; ++t) {
      v2f b = *(const v2f*)(wp + 32 * t);
      acc[0][t] = __builtin_amdgcn_wmma_f32_16x16x4_f32(false, a0, false, b, (short)0,
                                                        acc[0][t], false, true);
      acc[1][t] = __builtin_amdgcn_wmma_f32_16x16x4_f32(false, a1, false, b, (short)0,
                                                        acc[1][t], false, false);
    }
  }

  // C layout: VGPR r -> row (base + r + 8*(lane>=16)), col n0 + t*16 + (lane&15)
  #pragma unroll
  for (int rt = 0; rt < 2; ++rt) {
    const int mrow = rt * 16 + ((lane >> 4) << 3);
    float* outp = state + ((size_t)node * BATCH + b_off + mrow) * DIMN;
    #pragma unroll
    for (int t = 0; t < 4; ++t) {
      const int n = n0 + t * 16 + lm;
      const float bn = bias[n];
      #pragma unroll
      for (int r = 0; r < 8; ++r) {
        float x = acc[rt][t][r] + bn;
        outp[r * DIMN + n] = 1.0f / (1.0f + __expf(-x));
      }
    }
  }
}

// ---------------------------------------------------------------
// Kernel 4: emit root (node 0) state -> d_out [B, 256]
// ---------------------------------------------------------------
__global__ void copy_root_kernel(const float* __restrict__ state, float* __restrict__ out) {
  int i = blockIdx.x * blockDim.x + threadIdx.x;
  if (i < BATCH * DIMN) out[i] = state[i];   // node 0 occupies the first B*256 floats
}

// ---------------------------------------------------------------
extern "C" void kernel_launch(void* const* d_in, const int* in_sizes, int n_in,
                              void* d_out, int out_size, void* d_ws, size_t ws_size,
                              hipStream_t stream) {
  const int*   structure = (const int*)  d_in[0];  // [B, N, 2] int32 (batch-invariant)
  const float* features  = (const float*)d_in[1];  // [B, N, 256] f32
  const float* W_in      = (const float*)d_in[2];  // [256, 256]
  const float* b_in      = (const float*)d_in[3];  // [256]
  const float* W_inner   = (const float*)d_in[4];  // [768, 256]
  const float* b_inner   = (const float*)d_in[5];  // [256]

  // workspace: state (131 MB) | WtInner (512 KB) | WtLeaf (256 KB) | level meta (~8.5 KB)
  float* state        = (float*)d_ws;
  float* WtInner      = state + (size_t)NNODES * BATCH * DIMN;
  float* WtLeaf       = WtInner + 512 * 256;
  int*   level_count  = (int*)(WtLeaf + 256 * 256);
  int*   level_offset = level_count + MAXL;
  int*   level_nodes  = level_offset + MAXL;

  plan_kernel<<<1, 256, 0, stream>>>(structure, level_count, level_offset, level_nodes);
  prep_w_kernel<<<(768 * 256 + 255) / 256, 256, 0, stream>>>(W_inner, W_in, WtInner, WtLeaf);

  // Fixed launch sequence (graph-capture safe): level 0 = leaves (WtLeaf),
  // levels 1..MAXL-1 = inner nodes (WtInner). Blocks past the level's node
  // count exit immediately via the device-side count.
  for (int lvl = 0; lvl < MAXL; ++lvl) {
    tree_gemm_kernel<<<NNODES * 2, 128, 0, stream>>>(
        lvl, features,
        (lvl == 0) ? WtLeaf : WtInner,
        (lvl == 0) ? b_in : b_inner,
        structure, level_count, level_offset, level_nodes, state);
  }

  copy_root_kernel<<<(BATCH * DIMN + 255) / 256, 256, 0, stream>>>(state, (float*)d_out);
}